// EqxCGIteration_90993177133598
// MI455X (gfx1250) — compile-verified
//
#include <hip/hip_runtime.h>

typedef __attribute__((ext_vector_type(2))) float v2f;
typedef __attribute__((ext_vector_type(8))) float v8f;

__device__ __forceinline__ v8f wmma_f32_16x16x4(v2f a, v2f b, v8f c) {
  // 8 args: (neg_a, A, neg_b, B, c_mod, C, reuse_a, reuse_b)
  return __builtin_amdgcn_wmma_f32_16x16x4_f32(false, a, false, b, (short)0, c,
                                               false, false);
}

// One block handles ATOMS atoms of one l-channel.
// f : [n, D, D, K] row-major  -> rows m = (atom*DD + i*D + j), cols = channels
// out = f + ( tp( rmsnorm(f)*wr @ wi ) @ wo )
template <int D, int K, int ATOMS>
__global__ __launch_bounds__(256) void cg_iter_kernel(
    const float* __restrict__ fin, const float* __restrict__ wr,
    const float* __restrict__ wi, const float* __restrict__ wo,
    float* __restrict__ out, int n) {
  constexpr int DD = D * D;
  constexpr int M = ATOMS * DD;       // GEMM rows per block (multiple of 16)
  constexpr int MTILES = M / 16;
  constexpr int GTILES = K / 16;
  constexpr int NWAVES = 8;
  constexpr int NMT = (MTILES + NWAVES - 1) / NWAVES;  // m-tiles per wave
  constexpr int TPA = 256 / ATOMS;    // threads per atom in the reduction
  constexpr float INVSQRTD = (D == 1)   ? 1.0f
                             : (D == 3) ? 0.5773502691896258f
                             : (D == 5) ? 0.4472135954999579f
                                        : 0.3779644730092272f;

  __shared__ float sh_h[M * K];       // h tile (never hits HBM)
  __shared__ float sh_t[M * 16];      // tp(h) for one 16-channel K-block
  __shared__ float sh_wr[K];
  __shared__ float sh_red[256];
  __shared__ float sh_inv[ATOMS];

  const int tid = threadIdx.x;
  const int lane = tid & 31;
  const int wave = tid >> 5;
  const int col = lane & 15;          // A row / B col / C col
  const int koff = (lane >> 4) << 1;  // K-pair select: 0 or 2
  const int rbase = (lane >> 4) << 3; // C/D row base: 0 or 8

  const int atom_base = blockIdx.x * ATOMS;
  const long long grow0 = (long long)atom_base * DD;
  const long long mtot = (long long)n * DD;

  // ---- stage wr ----
  for (int t = tid; t < K; t += 256) sh_wr[t] = wr[t];

  // ---- phase 1: per-atom sum of squares -> inv (clamped, branch-free) ----
  {
    int a = tid / TPA;
    const int sub = tid % TPA;
    int aa = atom_base + a;
    if (aa >= n) aa = n - 1;  // clamp: garbage confined to OOB atoms
    const float* fa = fin + (long long)aa * (DD * K);
    float ss = 0.0f;
    for (int e = sub; e < DD * K; e += TPA) {
      float v = fa[e];
      ss += v * v;
    }
    sh_red[tid] = ss;
  }
  __syncthreads();
  if (tid < ATOMS) {
    float s = 0.0f;
#pragma unroll
    for (int q = 0; q < TPA; q++) s += sh_red[tid * TPA + q];
    sh_inv[tid] = rsqrtf(s * (1.0f / (float)(DD * K)) + 1e-6f);
  }
  __syncthreads();

  v8f zero;
#pragma unroll
  for (int q = 0; q < 8; q++) zero[q] = 0.0f;

  // ---- phase 2: GEMM1  h = (f * inv * wr) @ wi  -> sh_h ----
  // Each wave owns whole m-tiles; A fragment loaded once, reused for GTILES WMMAs.
  for (int mi = 0; mi < NMT; mi++) {
    const int mt = wave + mi * NWAVES;   // wave-uniform
    if (mt >= MTILES) break;
    const int m = mt * 16 + col;
    long long gr = grow0 + m;
    if (gr >= mtot) gr = mtot - 1;       // clamp instead of predicating loads
    const float scale = sh_inv[m / DD];
    const float* xr = fin + gr * (long long)K;

    v8f cacc[GTILES];
#pragma unroll
    for (int gt = 0; gt < GTILES; gt++) cacc[gt] = zero;

    for (int kb = 0; kb < K; kb += 4) {
      const v2f x2 = *(const v2f*)(xr + kb + koff);          // global_load_b64
      const v2f w2 = *(const v2f*)(sh_wr + kb + koff);       // ds_load_b64
      v2f av;
      av.x = x2.x * scale * w2.x;
      av.y = x2.y * scale * w2.y;
      const float* wrow0 = wi + (long long)(kb + koff) * K;
#pragma unroll
      for (int gt = 0; gt < GTILES; gt++) {
        v2f bv;
        bv.x = wrow0[gt * 16 + col];
        bv.y = wrow0[K + gt * 16 + col];
        cacc[gt] = wmma_f32_16x16x4(av, bv, cacc[gt]);
      }
    }
#pragma unroll
    for (int gt = 0; gt < GTILES; gt++)
#pragma unroll
      for (int v = 0; v < 8; v++)
        sh_h[(mt * 16 + rbase + v) * K + gt * 16 + col] = cacc[gt][v];
  }
  __syncthreads();

  // ---- phases 3/4: K-blocked tp + GEMM2 accumulation ----
  v8f acc[NMT][GTILES];
#pragma unroll
  for (int mi = 0; mi < NMT; mi++)
#pragma unroll
    for (int gt = 0; gt < GTILES; gt++) acc[mi][gt] = zero;

  for (int fb = 0; fb < GTILES; fb++) {
    // tp(h) for this 16-channel block (channel-diagonal contraction)
    for (int idx = tid; idx < M * 16; idx += 256) {
      const int m = idx >> 4;
      const int cch = idx & 15;
      const int a = m / DD;
      const int r = m % DD;
      const int ii = r / D;
      const int kk = r % D;
      const int g2 = fb * 16 + cch;
      float s = 0.0f;
#pragma unroll
      for (int j = 0; j < D; j++)
        s += sh_h[(a * DD + ii * D + j) * K + g2] *
             sh_h[(a * DD + j * D + kk) * K + g2];
      sh_t[m * 16 + cch] = s * INVSQRTD;
    }
    __syncthreads();

    // acc += t[:, fb-block] @ wo[fb-block, :], A fragment reused across gt
#pragma unroll
    for (int mi = 0; mi < NMT; mi++) {
      const int mt = wave + mi * NWAVES;  // wave-uniform branch
      if (mt < MTILES) {
#pragma unroll
        for (int kb = 0; kb < 16; kb += 4) {
          const v2f av = *(const v2f*)(sh_t + (mt * 16 + col) * 16 + kb + koff);
          const float* wrow0 = wo + (long long)(fb * 16 + kb + koff) * K;
#pragma unroll
          for (int gt = 0; gt < GTILES; gt++) {
            v2f bv;
            bv.x = wrow0[gt * 16 + col];
            bv.y = wrow0[K + gt * 16 + col];
            acc[mi][gt] = wmma_f32_16x16x4(av, bv, acc[mi][gt]);
          }
        }
      }
    }
    __syncthreads();
  }

  // ---- epilogue: out = f + acc ----
#pragma unroll
  for (int mi = 0; mi < NMT; mi++) {
    const int mt = wave + mi * NWAVES;
    if (mt < MTILES) {
#pragma unroll
      for (int gt = 0; gt < GTILES; gt++) {
#pragma unroll
        for (int v = 0; v < 8; v++) {
          const long long gr = grow0 + mt * 16 + rbase + v;
          if (gr < mtot) {
            const long long gi = gr * K + gt * 16 + col;
            out[gi] = fin[gi] + acc[mi][gt][v];
          }
        }
      }
    }
  }
}

extern "C" void kernel_launch(void* const* d_in, const int* in_sizes, int n_in,
                              void* d_out, int out_size, void* d_ws,
                              size_t ws_size, hipStream_t stream) {
  (void)n_in;
  (void)out_size;
  (void)d_ws;
  (void)ws_size;
  // setup_inputs() dict order: f0,wr0,wi0,wo0, f1,wr1,wi1,wo1, ...
  const float* f0 = (const float*)d_in[0];
  const float* wr0 = (const float*)d_in[1];
  const float* wi0 = (const float*)d_in[2];
  const float* wo0 = (const float*)d_in[3];
  const float* f1 = (const float*)d_in[4];
  const float* wr1 = (const float*)d_in[5];
  const float* wi1 = (const float*)d_in[6];
  const float* wo1 = (const float*)d_in[7];
  const float* f2 = (const float*)d_in[8];
  const float* wr2 = (const float*)d_in[9];
  const float* wi2 = (const float*)d_in[10];
  const float* wo2 = (const float*)d_in[11];
  const float* f3 = (const float*)d_in[12];
  const float* wr3 = (const float*)d_in[13];
  const float* wi3 = (const float*)d_in[14];
  const float* wo3 = (const float*)d_in[15];

  float* out = (float*)d_out;
  const int n = in_sizes[0] / 256;  // N (f0 is [N,1,1,256])
  const long long o1 = (long long)n * 1 * 1 * 256;
  const long long o2 = o1 + (long long)n * 3 * 3 * 128;
  const long long o3 = o2 + (long long)n * 5 * 5 * 64;

  const dim3 blk(256);
  const int g0 = (n + 127) / 128;
  const int g16 = (n + 15) / 16;
  cg_iter_kernel<1, 256, 128><<<g0, blk, 0, stream>>>(f0, wr0, wi0, wo0, out, n);
  cg_iter_kernel<3, 128, 16><<<g16, blk, 0, stream>>>(f1, wr1, wi1, wo1, out + o1, n);
  cg_iter_kernel<5, 64, 16><<<g16, blk, 0, stream>>>(f2, wr2, wi2, wo2, out + o2, n);
  cg_iter_kernel<7, 32, 16><<<g16, blk, 0, stream>>>(f3, wr3, wi3, wo3, out + o3, n);
}